// VQ_VAE_ema_41729902248239
// MI455X (gfx1250) — compile-verified
//
#include <hip/hip_runtime.h>
#include <hip/hip_bf16.h>

// ---------------- problem constants ----------------
#define NB   32          // batch
#define CC   64          // channels
#define PP   4096        // H*W
#define SS   512         // codebook size
#define NP   (NB*PP)     // tokens = 131072
#define NELT (NB*CC*PP)  // 8388608

#define LDS_STRIDE 136   // 128 codes + 8 pad dwords (disjoint 16-bank halves)

typedef float v8f __attribute__((ext_vector_type(8)));
typedef float v2f __attribute__((ext_vector_type(2)));

// ---------------- prep: transpose/scale codebook, norms, zero accumulators ----
__global__ void vq_prep(const float* __restrict__ cb,
                        float* __restrict__ wsT,      // [64][512] = -2*cb^T
                        float* __restrict__ wsNorms,  // [512]
                        float* __restrict__ wsCounts, // [512]
                        float* __restrict__ wsDw,     // [512*64]
                        float* __restrict__ wsLoss) { // [1]
    int s = blockIdx.x;      // code
    int c = threadIdx.x;     // channel
    float v = cb[s * CC + c];
    wsT[c * SS + s] = -2.0f * v;
    wsDw[s * CC + c] = 0.0f;
    __shared__ float red[CC];
    red[c] = v * v;
    __syncthreads();
    for (int st = 32; st > 0; st >>= 1) {
        if (c < st) red[c] += red[c + st];
        __syncthreads();
    }
    if (c == 0) {
        wsNorms[s] = red[0];
        wsCounts[s] = 0.0f;
        if (s == 0) wsLoss[0] = 0.0f;
    }
}

// ---------------- main: WMMA distances + argmin + gather/scatter ------------
__global__ void __launch_bounds__(256)
vq_main(const float* __restrict__ x,
        const float* __restrict__ cb,        // original codebook (for xq gather)
        const float* __restrict__ wsT,       // -2*cb^T  [64][512]
        const float* __restrict__ wsNorms,   // ||c||^2  [512]
        float* __restrict__ wsCounts,
        float* __restrict__ wsDw,
        float* __restrict__ wsLoss,
        float* __restrict__ oIdx,            // [NP] (argmin as float)
        float* __restrict__ oX) {            // [N,C,P] quantized output
    __shared__ float ldsB[CC * LDS_STRIDE];  // 64 x (128+pad) chunk of -2*cb^T
    __shared__ float sBestV[8][16];
    __shared__ int   sBestI[8][16];

    const int tid  = threadIdx.x;
    const int wave = tid >> 5;
    const int lane = tid & 31;
    const int row  = lane & 15;   // token row (A:M, D:N-col owner), code col idx
    const int hb   = lane >> 4;   // half-wave selects K pair / M offset

    // 8 waves per block, one 16-token tile each
    const int tile = blockIdx.x * 8 + wave;
    const int t0   = tile * 16;
    const int n    = t0 >> 12;           // / 4096
    const int p0   = t0 & 4095;

    const float* xbase = x + ((size_t)n * CC) * PP + p0 + row;

    // ---- load A tile: 16 tokens x 64 channels in WMMA f32 16x16x4 layout ----
    float a[32];
#pragma unroll
    for (int kk = 0; kk < 16; ++kk) {
        int c = 4 * kk + 2 * hb;
        a[2 * kk]     = xbase[(size_t)c * PP];
        a[2 * kk + 1] = xbase[(size_t)(c + 1) * PP];
    }
    float ssq = 0.0f;
#pragma unroll
    for (int i = 0; i < 32; ++i) ssq += a[i] * a[i];
    float tnorm = ssq + __shfl_xor(ssq, 16, 32);   // full ||t_row||^2 in all lanes

    float bestv[8];
    int   besti[8];
#pragma unroll
    for (int v = 0; v < 8; ++v) { bestv[v] = 3.4e38f; besti[v] = 0; }

    // ---- sweep codebook in 4 LDS chunks of 128 codes ----
    for (int chunk = 0; chunk < 4; ++chunk) {
        for (int i = tid; i < CC * 128; i += 256) {
            int c  = i >> 7;
            int sl = i & 127;
            ldsB[c * LDS_STRIDE + sl] = wsT[c * SS + chunk * 128 + sl];
        }
        __syncthreads();

        for (int jj = 0; jj < 8; ++jj) {
            const int codeBase = chunk * 128 + jj * 16;
            const float vn = wsNorms[codeBase + row];   // init acc with ||c||^2
            v8f acc = {vn, vn, vn, vn, vn, vn, vn, vn};
#pragma unroll
            for (int kk = 0; kk < 16; ++kk) {
                int c = 4 * kk + 2 * hb;
                v2f av = {a[2 * kk], a[2 * kk + 1]};
                v2f bv = {ldsB[c * LDS_STRIDE + jj * 16 + row],
                          ldsB[(c + 1) * LDS_STRIDE + jj * 16 + row]};
                acc = __builtin_amdgcn_wmma_f32_16x16x4_f32(
                        false, av, false, bv, (short)0, acc, false, false);
            }
            const int cand = codeBase + row;
#pragma unroll
            for (int v = 0; v < 8; ++v) {
                if (acc[v] < bestv[v]) { bestv[v] = acc[v]; besti[v] = cand; }
            }
        }
        __syncthreads();
    }

    // ---- butterfly min-reduce over 16-lane groups (N dimension) ----
#pragma unroll
    for (int v = 0; v < 8; ++v) {
#pragma unroll
        for (int m = 1; m < 16; m <<= 1) {
            float ov = __shfl_xor(bestv[v], m, 32);
            int   oi = __shfl_xor(besti[v], m, 32);
            if (ov < bestv[v] || (ov == bestv[v] && oi < besti[v])) {
                bestv[v] = ov; besti[v] = oi;
            }
        }
    }
    // lanes 0..15 hold rows 0..7; lanes 16..31 hold rows 8..15
    if (row == 0) {
#pragma unroll
        for (int v = 0; v < 8; ++v) {
            sBestV[wave][v + 8 * hb] = bestv[v];
            sBestI[wave][v + 8 * hb] = besti[v];
        }
    }
    __syncthreads();

    const int   bidx = sBestI[wave][row];
    const float bmin = sBestV[wave][row];

    if (hb == 0) {
        oIdx[t0 + row] = (float)bidx;
        atomicAdd(&wsCounts[bidx], 1.0f);
        float lossv = tnorm + bmin;            // ||t||^2 + (||c||^2 - 2 t.c)
#pragma unroll
        for (int m = 1; m < 16; m <<= 1) lossv += __shfl_xor(lossv, m, 32);
        if (lane == 0) atomicAdd(wsLoss, lossv);
    }

    // ---- write xq and scatter dw (reuse resident A registers) ----
    const float* cbr = cb + (size_t)bidx * CC;
    float* dwr = wsDw + (size_t)bidx * CC;
#pragma unroll
    for (int kk = 0; kk < 16; ++kk) {
        int c = 4 * kk + 2 * hb;
        size_t o = ((size_t)(n * CC + c)) * PP + p0 + row;
        oX[o]      = cbr[c];
        oX[o + PP] = cbr[c + 1];
        atomicAdd(&dwr[c],     a[2 * kk]);
        atomicAdd(&dwr[c + 1], a[2 * kk + 1]);
    }
}

// ---------------- finalize: EMA update, bias correction, loss ---------------
__global__ void vq_finalize(const float* __restrict__ ema_w,
                            const float* __restrict__ cs_old,
                            const int*   __restrict__ steps,
                            const float* __restrict__ wsCounts,
                            const float* __restrict__ wsDw,
                            const float* __restrict__ wsLoss,
                            float* __restrict__ oLoss,
                            float* __restrict__ oCb,
                            float* __restrict__ oCs,
                            float* __restrict__ oEw) {
    const int s = threadIdx.x;       // one block of 512
    const float bias = 1.0f - powf(0.99f, (float)steps[0]);
    const float csn  = 0.99f * cs_old[s] + 0.01f * wsCounts[s];
    oCs[s] = csn;
    const float csbc = csn / bias;

    __shared__ float red[SS];
    red[s] = csbc;
    __syncthreads();
    for (int st = 256; st > 0; st >>= 1) {
        if (s < st) red[s] += red[s + st];
        __syncthreads();
    }
    const float nsum = red[0];
    const float upd = (csbc + 1e-7f) / (nsum + SS * 1e-7f) * nsum;

    for (int c = 0; c < CC; ++c) {
        float ew = 0.99f * ema_w[s * CC + c] + 0.01f * wsDw[s * CC + c];
        oEw[s * CC + c] = ew;
        oCb[s * CC + c] = (ew / bias) / upd;
    }
    if (s == 0) oLoss[0] = wsLoss[0] / (float)NELT;
}

// ---------------- host launch ----------------
extern "C" void kernel_launch(void* const* d_in, const int* in_sizes, int n_in,
                              void* d_out, int out_size, void* d_ws, size_t ws_size,
                              hipStream_t stream) {
    const float* x        = (const float*)d_in[0];
    const float* codebook = (const float*)d_in[1];
    const float* ema_w    = (const float*)d_in[2];
    const float* cs_old   = (const float*)d_in[3];
    const int*   steps    = (const int*)d_in[4];

    float* wsT      = (float*)d_ws;        // 32768
    float* wsNorms  = wsT + SS * CC;       // 512
    float* wsCounts = wsNorms + SS;        // 512
    float* wsDw     = wsCounts + SS;       // 32768
    float* wsLoss   = wsDw + SS * CC;      // 1

    float* oLoss = (float*)d_out;
    float* oIdx  = oLoss + 1;
    float* oX    = oIdx + NP;
    float* oCb   = oX + NELT;
    float* oCs   = oCb + SS * CC;
    float* oEw   = oCs + SS;

    vq_prep<<<SS, CC, 0, stream>>>(codebook, wsT, wsNorms, wsCounts, wsDw, wsLoss);
    vq_main<<<NP / 128, 256, 0, stream>>>(x, codebook, wsT, wsNorms,
                                          wsCounts, wsDw, wsLoss, oIdx, oX);
    vq_finalize<<<1, SS, 0, stream>>>(ema_w, cs_old, steps,
                                      wsCounts, wsDw, wsLoss,
                                      oLoss, oCb, oCs, oEw);
}